// GATTriple_55224689492117
// MI455X (gfx1250) — compile-verified
//
#include <hip/hip_runtime.h>

typedef __attribute__((ext_vector_type(2))) float v2f;
typedef __attribute__((ext_vector_type(8))) float v8f;

#define NEG_SLOPE 0.2f
#define KC 32   // K-chunk staged in LDS per iteration (all Ks here are multiples of 32)

#if __has_builtin(__builtin_amdgcn_global_load_async_to_lds_b32)
#define HAVE_ASYNC_LDS 1
typedef __attribute__((address_space(1))) int gint;   // global (HIP prints as __device__)
typedef __attribute__((address_space(3))) int lint;   // LDS
#endif

__device__ __forceinline__ float leaky(float x) {
    return x >= 0.0f ? x : NEG_SLOPE * x;
}

__device__ __forceinline__ void wait_async_then_barrier() {
#ifdef HAVE_ASYNC_LDS
#if __has_builtin(__builtin_amdgcn_s_wait_asynccnt)
    __builtin_amdgcn_s_wait_asynccnt(0);
#else
    asm volatile("s_wait_asynccnt 0x0" ::: "memory");
#endif
#endif
    __syncthreads();
}

// -----------------------------------------------------------------------------
// C[M x 32] = A @ B[K x 32] with V_WMMA_F32_16X16X4_F32.
// TRANSA == false: A element (m,k) = A[m*lda + k]   (row-major features)
// TRANSA == true : A element (m,k) = A[k*lda + m]   (K-major softmax weights)
// grid.x = M/16, blockDim = 64 (2 waves; wave w owns output cols 16w..16w+15).
// B chunk (32x32) staged in LDS; A chunk (32x16) staged too on the TRANSA path
// (coalesced over m). K must be a multiple of KC.
//
// Per-lane fragment layouts (ISA 7.12.2, 32-bit):
//   A 16x4 :  lane L: row = L%16, VGPR0/1 = K + (L<16 ? 0 : 2) + {0,1}
//   B 4x16 :  lane L: col = L%16, VGPR v  = row K + (L<16 ? 0 : 2) + v
//   C 16x16:  lane L: col = L%16, VGPR v  = row v + (L<16 ? 0 : 8)
// -----------------------------------------------------------------------------
template <bool TRANSA>
__global__ void wmma_gemm32(const float* __restrict__ A,
                            const float* __restrict__ B,
                            float* __restrict__ C,
                            int K, int lda) {
    __shared__ float Bs[KC * 32];          // [kk][n]
    __shared__ float As[KC * 16];          // [kk][mm] (TRANSA only)
    const int tid  = threadIdx.x;
    const int lane = tid & 31;
    const int wave = tid >> 5;
    const int m0   = blockIdx.x * 16;
    const int n0   = wave * 16;
    const int mr   = lane & 15;            // row/col within tile for this lane
    const int ks   = (lane < 16) ? 0 : 2;  // K sub-offset per lane half

    v8f acc = {0.f, 0.f, 0.f, 0.f, 0.f, 0.f, 0.f, 0.f};

    for (int k0 = 0; k0 < K; k0 += KC) {
        // ---- stage B[k0..k0+KC) x 32 (1024 floats, 16 per thread) ----
#pragma unroll
        for (int t = tid; t < KC * 32; t += 64) {
            const float* src = &B[(k0 + (t >> 5)) * 32 + (t & 31)];
#ifdef HAVE_ASYNC_LDS
            __builtin_amdgcn_global_load_async_to_lds_b32(
                (gint*)src, (lint*)&Bs[t], 0, 0);
#else
            Bs[t] = *src;
#endif
        }
        // ---- stage A[k0..k0+KC) x 16 for the K-major path (coalesced in m) ----
        if (TRANSA) {
#pragma unroll
            for (int t = tid; t < KC * 16; t += 64) {
                const float* src = &A[(k0 + (t >> 4)) * lda + m0 + (t & 15)];
#ifdef HAVE_ASYNC_LDS
                __builtin_amdgcn_global_load_async_to_lds_b32(
                    (gint*)src, (lint*)&As[t], 0, 0);
#else
                As[t] = *src;
#endif
            }
        }
        wait_async_then_barrier();
        if (tid == 0 && (k0 + KC) < K)
            __builtin_prefetch(&B[(k0 + KC) * 32], 0, 1);   // global_prefetch_b8

        // ---- 8 fully unrolled WMMAs over the chunk ----
#pragma unroll
        for (int kk = 0; kk < KC; kk += 4) {
            v2f a;
            if (TRANSA) {
                a.x = As[(kk + ks    ) * 16 + mr];
                a.y = As[(kk + ks + 1) * 16 + mr];
            } else {
                a = *reinterpret_cast<const v2f*>(A + (m0 + mr) * lda + k0 + kk + ks);
            }
            v2f b;   // this wave's B tile spans columns n0..n0+15
            b.x = Bs[(kk + ks    ) * 32 + n0 + mr];
            b.y = Bs[(kk + ks + 1) * 32 + n0 + mr];
            acc = __builtin_amdgcn_wmma_f32_16x16x4_f32(
                false, a, false, b, (short)0, acc, false, false);
        }
        __syncthreads();
    }

    const int rbase = m0 + ((lane < 16) ? 0 : 8);
#pragma unroll
    for (int v = 0; v < 8; ++v)
        C[(rbase + v) * 32 + n0 + mr] = acc[v];
}

// out[r] = dot(H[r, 0:32], a[0:32])  -- one wave per row
__global__ void rowdot(const float* __restrict__ H, const float* __restrict__ a,
                       float* __restrict__ out, int R) {
    const int lane = threadIdx.x & 31;
    const int row  = blockIdx.x * 8 + (threadIdx.x >> 5);
    if (row >= R) return;
    float v = H[row * 32 + lane] * a[lane];
#pragma unroll
    for (int off = 16; off > 0; off >>= 1) v += __shfl_xor(v, off, 32);
    if (lane == 0) out[row] = v;
}

// c[0] = dot(W_e1, a_e1), c[1] = dot(W_e2, a_e2)  -- two waves
__global__ void edge_const(const float* __restrict__ W_e1, const float* __restrict__ a_e1,
                           const float* __restrict__ W_e2, const float* __restrict__ a_e2,
                           float* __restrict__ c) {
    const int lane = threadIdx.x & 31;
    const int wave = threadIdx.x >> 5;
    const float* W = wave ? W_e2 : W_e1;
    const float* A = wave ? a_e2 : a_e1;
    float v = W[lane] * A[lane];
#pragma unroll
    for (int off = 16; off > 0; off >>= 1) v += __shfl_xor(v, off, 32);
    if (lane == 0) c[wave] = v;
}

// Column-wise masked softmax over source dim (rows), two-pass, thread-per-column.
// Stores unnormalized weights P[i*NJ + j], per-column sum(w) and sum(w*edge).
__global__ void col_softmax(const float* __restrict__ edge,
                            const float* __restrict__ el,
                            const float* __restrict__ er,
                            const float* __restrict__ cptr,
                            float* __restrict__ P,
                            float* __restrict__ sumw,
                            float* __restrict__ sedge,
                            int NI, int NJ) {
    const int j = blockIdx.x * blockDim.x + threadIdx.x;
    if (j >= NJ) return;
    const float c   = cptr[0];
    const float erj = er[j];

    float m = -__builtin_inff();
#pragma unroll 4
    for (int i = 0; i < NI; ++i) {
        float e = edge[i * NJ + j];
        if (e != 0.0f) m = fmaxf(m, leaky(el[i] + c * e + erj));
    }
    float sum = 0.0f, se = 0.0f;
#pragma unroll 4
    for (int i = 0; i < NI; ++i) {
        float e = edge[i * NJ + j];
        float w = 0.0f;
        if (e != 0.0f) w = __expf(leaky(el[i] + c * e + erj) - m);
        P[i * NJ + j] = w;
        sum += w;
        se  += w * e;
    }
    sumw[j]  = sum;
    sedge[j] = se;
}

// out[j,n] = sigmoid((Msg[j,n] + sedge[j]*W_e[n]) / sumw[j] + Hnode[j,n])
__global__ void combine(const float* __restrict__ Msg,
                        const float* __restrict__ sedge,
                        const float* __restrict__ sumw,
                        const float* __restrict__ W_e,
                        const float* __restrict__ Hnode,
                        float* __restrict__ out, int NJ) {
    const int t = blockIdx.x * blockDim.x + threadIdx.x;
    if (t >= NJ * 32) return;
    const int j = t >> 5, n = t & 31;
    const float msg = (Msg[t] + sedge[j] * W_e[n]) / sumw[j];
    out[t] = 1.0f / (1.0f + __expf(-(msg + Hnode[t])));
}

extern "C" void kernel_launch(void* const* d_in, const int* in_sizes, int n_in,
                              void* d_out, int out_size, void* d_ws, size_t ws_size,
                              hipStream_t stream) {
    const float* op_feat   = (const float*)d_in[0];
    const float* agv_feat  = (const float*)d_in[1];
    const float* mach_feat = (const float*)d_in[2];
    const float* edge1     = (const float*)d_in[3];   // [4096, 512]
    const float* edge2     = (const float*)d_in[4];   // [512, 128]
    const float* W_op   = (const float*)d_in[5];
    const float* W_agv  = (const float*)d_in[6];
    const float* W_mach = (const float*)d_in[7];
    const float* W_e1   = (const float*)d_in[8];
    const float* W_e2   = (const float*)d_in[9];
    const float* a_op   = (const float*)d_in[10];
    const float* a_agv  = (const float*)d_in[11];
    const float* a_mach = (const float*)d_in[12];
    const float* a_e1   = (const float*)d_in[13];
    const float* a_e2   = (const float*)d_in[14];
    float* out = (float*)d_out;

    const int N_OP = 4096, N_AGV = 512, N_MACH = 128;
    const int IN_OP = 64, IN_AGV = 32, IN_MACH = 32;

    float* ws      = (float*)d_ws;
    float* h_op    = ws;  ws += N_OP * 32;
    float* h_agv   = ws;  ws += N_AGV * 32;
    float* h_mach  = ws;  ws += N_MACH * 32;
    float* el_op   = ws;  ws += N_OP;
    float* er_agv  = ws;  ws += N_AGV;     // also el2 for block 2
    float* er_mach = ws;  ws += N_MACH;
    float* cbuf    = ws;  ws += 2;
    float* P1      = ws;  ws += N_OP * N_AGV;
    float* sum1    = ws;  ws += N_AGV;
    float* se1     = ws;  ws += N_AGV;
    float* P2      = ws;  ws += N_AGV * N_MACH;
    float* sum2    = ws;  ws += N_MACH;
    float* se2     = ws;  ws += N_MACH;
    float* M1      = ws;  ws += N_AGV * 32;
    float* M2      = ws;  ws += N_MACH * 32;

    // scalar edge-projection constants (on device: no host readback in capture)
    edge_const<<<1, 64, 0, stream>>>(W_e1, a_e1, W_e2, a_e2, cbuf);

    // node projections (f32 WMMA)
    wmma_gemm32<false><<<N_OP   / 16, 64, 0, stream>>>(op_feat,   W_op,   h_op,   IN_OP,   IN_OP);
    wmma_gemm32<false><<<N_AGV  / 16, 64, 0, stream>>>(agv_feat,  W_agv,  h_agv,  IN_AGV,  IN_AGV);
    wmma_gemm32<false><<<N_MACH / 16, 64, 0, stream>>>(mach_feat, W_mach, h_mach, IN_MACH, IN_MACH);

    // attention logits per node
    rowdot<<<N_OP   / 8, 256, 0, stream>>>(h_op,   a_op,   el_op,   N_OP);
    rowdot<<<N_AGV  / 8, 256, 0, stream>>>(h_agv,  a_agv,  er_agv,  N_AGV);
    rowdot<<<N_MACH / 8, 256, 0, stream>>>(h_mach, a_mach, er_mach, N_MACH);

    // ---- block 1: op -> agv ----
    col_softmax<<<(N_AGV + 127) / 128, 128, 0, stream>>>(
        edge1, el_op, er_agv, cbuf + 0, P1, sum1, se1, N_OP, N_AGV);
    // agv_msg (unnormalized) = P1^T @ h_op   [512 x 4096 x 32]
    wmma_gemm32<true><<<N_AGV / 16, 64, 0, stream>>>(P1, h_op, M1, N_OP, N_AGV);
    combine<<<(N_AGV * 32) / 256, 256, 0, stream>>>(M1, se1, sum1, W_e1, h_agv, out, N_AGV);

    // ---- block 2: agv -> mach (el2 == er_agv since a_agv reused) ----
    col_softmax<<<1, 128, 0, stream>>>(
        edge2, er_agv, er_mach, cbuf + 1, P2, sum2, se2, N_AGV, N_MACH);
    wmma_gemm32<true><<<N_MACH / 16, 64, 0, stream>>>(P2, h_agv, M2, N_AGV, N_MACH);
    combine<<<(N_MACH * 32) / 256, 256, 0, stream>>>(M2, se2, sum2, W_e2, h_mach,
                                                     out + N_AGV * 32, N_MACH);
}